// IoU_10067403341969
// MI455X (gfx1250) — compile-verified
//
#include <hip/hip_runtime.h>
#include <math.h>

#define BCE_COEFF 0.2f
#define WAVE 32
#define BLOCK 256
#define GRID 1024   // fixed grid -> fixed accumulation order -> deterministic

// Fast sigmoid using gfx1250 hardware transcendentals: v_exp_f32 + v_rcp_f32.
__device__ __forceinline__ float fast_sigmoid(float x) {
    return __builtin_amdgcn_rcpf(1.0f + __expf(-x));
}

// One row of the fused IoU + BCE loss.
// a0..a4 = preds row, g0..g4 = target row.
__device__ __forceinline__ void row_terms(float a0, float a1, float a2, float a3, float a4,
                                          float g0, float g1, float g2, float g3, float g4,
                                          float& bce_acc, float& iou_acc) {
    // s = sigmoid(preds[:,1:5]); t = target[:,1:5]
    const float s0 = fast_sigmoid(a1);
    const float s1 = fast_sigmoid(a2);
    const float s2 = fast_sigmoid(a3);
    const float s3 = fast_sigmoid(a4);
    const float t0 = g1, t1 = g2, t2 = g3, t3 = g4;

    const float pred_area   = fabsf(s2 - s0) * fabsf(s3 - s1);
    const float target_area = (t2 - t0) * (t3 - t1);

    const float x_p1 = fminf(s2, s0);
    const float x_p2 = fmaxf(s2, s0);
    const float y_p1 = fminf(s1, s3);
    const float y_p2 = fmaxf(s1, s3);

    const float x1 = fmaxf(x_p1, t0);
    const float x2 = fminf(x_p2, t2);
    const float y1 = fmaxf(y_p1, t1);
    const float y2 = fminf(y_p2, t3);

    const float overlap = (x2 - x1) * (y2 - y1);
    const float denom   = target_area + pred_area - overlap;
    const float iou     = (overlap < 0.0f) ? 0.0f : overlap * __builtin_amdgcn_rcpf(denom);
    iou_acc += iou;

    // Stable BCE-with-logits on column 0: max(l,0) - l*y + log1p(exp(-|l|))
    const float l = a0, y = g0;
    const float e = __expf(-fabsf(l));
    bce_acc += fmaxf(l, 0.0f) - l * y + __logf(1.0f + e);
}

// Process one 4-row chunk (80 B = 5x float4 per array, 16B aligned).
__device__ __forceinline__ void chunk4(const float* __restrict__ preds,
                                       const float* __restrict__ target,
                                       size_t chunk, float& bce, float& iou) {
    const float4* __restrict__ pa = reinterpret_cast<const float4*>(preds  + chunk * 20);
    const float4* __restrict__ ga = reinterpret_cast<const float4*>(target + chunk * 20);
    float a[20], g[20];
    float4* av = reinterpret_cast<float4*>(a);
    float4* gv = reinterpret_cast<float4*>(g);
    #pragma unroll
    for (int k = 0; k < 5; ++k) { av[k] = pa[k]; }   // 5x global_load_b128
    #pragma unroll
    for (int k = 0; k < 5; ++k) { gv[k] = ga[k]; }   // 5x global_load_b128
    #pragma unroll
    for (int r = 0; r < 4; ++r) {
        row_terms(a[r*5+0], a[r*5+1], a[r*5+2], a[r*5+3], a[r*5+4],
                  g[r*5+0], g[r*5+1], g[r*5+2], g[r*5+3], g[r*5+4],
                  bce, iou);
    }
}

// Kernel 1: fixed-grid grid-stride loop over 4-row chunks with global_prefetch_b8
// software pipelining; wave32 + LDS reduction; one (bce, iou) partial per block.
__global__ void __launch_bounds__(BLOCK)
iou_bce_partial(const float* __restrict__ preds,
                const float* __restrict__ target,
                float* __restrict__ partial,
                int nChunks, int nRows) {
    const int tid    = blockIdx.x * blockDim.x + threadIdx.x;
    const int stride = gridDim.x * blockDim.x;

    float bce = 0.0f, iou = 0.0f;

    for (int c = tid; c < nChunks; c += stride) {
        // Prefetch next iteration's chunk into cache (gfx1250 global_prefetch_b8).
        const int cn = c + stride;
        if (cn < nChunks) {
            __builtin_prefetch(preds  + (size_t)cn * 20, 0, 3);
            __builtin_prefetch(target + (size_t)cn * 20, 0, 3);
        }
        chunk4(preds, target, (size_t)c, bce, iou);
    }

    // Scalar tail for nRows % 4 != 0 (not taken for N = 4194304).
    const int r = nChunks * 4 + tid;
    if (r < nRows) {
        const float* a = preds  + (size_t)r * 5;
        const float* g = target + (size_t)r * 5;
        row_terms(a[0], a[1], a[2], a[3], a[4],
                  g[0], g[1], g[2], g[3], g[4], bce, iou);
    }

    // wave32 butterfly reduction (CDNA5 is wave32-only)
    #pragma unroll
    for (int m = WAVE / 2; m > 0; m >>= 1) {
        bce += __shfl_xor(bce, m, WAVE);
        iou += __shfl_xor(iou, m, WAVE);
    }

    __shared__ float sb[BLOCK / WAVE];
    __shared__ float si[BLOCK / WAVE];
    const int lane = threadIdx.x & (WAVE - 1);
    const int wv   = threadIdx.x / WAVE;
    if (lane == 0) { sb[wv] = bce; si[wv] = iou; }
    __syncthreads();

    if (threadIdx.x == 0) {
        float b = 0.0f, i = 0.0f;
        #pragma unroll
        for (int w = 0; w < BLOCK / WAVE; ++w) { b += sb[w]; i += si[w]; }
        partial[2 * (size_t)blockIdx.x + 0] = b;
        partial[2 * (size_t)blockIdx.x + 1] = i;
    }
}

// Kernel 2: deterministic single-block reduction of block partials -> scalar loss.
__global__ void __launch_bounds__(BLOCK)
iou_bce_final(const float* __restrict__ partial,
              int nPartials,
              float* __restrict__ out,
              float invN) {
    float b = 0.0f, i = 0.0f;
    for (int k = threadIdx.x; k < nPartials; k += blockDim.x) {
        b += partial[2 * (size_t)k + 0];
        i += partial[2 * (size_t)k + 1];
    }
    #pragma unroll
    for (int m = WAVE / 2; m > 0; m >>= 1) {
        b += __shfl_xor(b, m, WAVE);
        i += __shfl_xor(i, m, WAVE);
    }
    __shared__ float sb[BLOCK / WAVE];
    __shared__ float si[BLOCK / WAVE];
    const int lane = threadIdx.x & (WAVE - 1);
    const int wv   = threadIdx.x / WAVE;
    if (lane == 0) { sb[wv] = b; si[wv] = i; }
    __syncthreads();

    if (threadIdx.x == 0) {
        float tb = 0.0f, ti = 0.0f;
        #pragma unroll
        for (int w = 0; w < BLOCK / WAVE; ++w) { tb += sb[w]; ti += si[w]; }
        // loss = BCE_COEFF * mean(bce) + (1 - mean(iou))
        out[0] = BCE_COEFF * (tb * invN) + (1.0f - ti * invN);
    }
}

extern "C" void kernel_launch(void* const* d_in, const int* in_sizes, int n_in,
                              void* d_out, int out_size, void* d_ws, size_t ws_size,
                              hipStream_t stream) {
    const float* preds  = (const float*)d_in[0];
    const float* target = (const float*)d_in[1];
    float* out = (float*)d_out;
    float* ws  = (float*)d_ws;

    const int nRows   = in_sizes[0] / 5;   // 4194304
    const int nChunks = nRows / 4;         // 1048576 four-row chunks

    iou_bce_partial<<<GRID, BLOCK, 0, stream>>>(preds, target, ws, nChunks, nRows);
    iou_bce_final<<<1, BLOCK, 0, stream>>>(ws, GRID, out, 1.0f / (float)nRows);
}